// TF_physical_poly_field_13365938225470
// MI455X (gfx1250) — compile-verified
//
#include <hip/hip_runtime.h>
#include <math.h>

#ifndef M_PI
#define M_PI 3.14159265358979323846
#endif

typedef float v8f __attribute__((ext_vector_type(8)));
typedef float v2f __attribute__((ext_vector_type(2)));

#define OPD    256
#define NB     20
#define NZT    66
#define NZP    45
#define NP2    6
#define NP5    21
#define NK     87          // 66 zernike + 21 S-map coefficients
#define CROP   96
#define OD     32
#define BATCHN 32
#define NSTARS 2000

// workspace layout (float offsets)
#define OFF_ZC   0
#define OFF_OPD  4096
#define OFF_WR   (OFF_OPD + BATCHN*OPD*OPD)
#define OFF_WI   (OFF_WR + NB*CROP*OPD)
#define OFF_TR   (OFF_WI + NB*CROP*OPD)
#define OFF_TI   (OFF_TR + BATCHN*CROP*OPD)
#define OFF_PS   (OFF_TI + BATCHN*CROP*OPD)

struct PN { int pn[NB]; };

// ---------------------------------------------------------------------------
// K0: per-star combined coefficient vector zc[b][0..86]
//     [0..65] = padded param zernikes + prior (via exact position match)
//     [66..86] = pm_np^T @ alpha_mat
// ---------------------------------------------------------------------------
__global__ void k0_zcomb(const float* __restrict__ positions,
                         const float* __restrict__ coeff_mat,
                         const float* __restrict__ alpha_mat,
                         const float* __restrict__ obs_pos,
                         const float* __restrict__ zks_prior,
                         float* __restrict__ ws) {
  int b = threadIdx.x;
  if (b >= BATCHN) return;
  float px = positions[2*b], py = positions[2*b+1];
  int idx = 0;
  for (int i = 0; i < NSTARS; ++i) {
    if (obs_pos[2*i] == px && obs_pos[2*i+1] == py) { idx = i; break; }
  }
  float x = px * (2.0f/1000.0f) - 1.0f;
  float y = py * (2.0f/1000.0f) - 1.0f;
  float pm[NP5];
  int c = 0;
  for (int d = 0; d <= 5; ++d)
    for (int p = 0; p <= d; ++p) {
      float v = 1.0f;
      for (int q = 0; q < d - p; ++q) v *= x;
      for (int q = 0; q < p; ++q)     v *= y;
      pm[c++] = v;
    }
  float* zc = ws + OFF_ZC + b*NK;
  for (int k = 0; k < NZT; ++k) {
    float v = 0.0f;
    if (k < NZP)
      for (int q = 0; q < NP2; ++q) v += coeff_mat[k*NP2 + q] * pm[q];
    zc[k] = v + zks_prior[idx*NZT + k];
  }
  for (int q = 0; q < NP5; ++q) {
    float v = 0.0f;
    for (int p = 0; p < NP5; ++p) v += pm[p] * alpha_mat[p*NP5 + q];
    zc[NZT + q] = v;
  }
}

// ---------------------------------------------------------------------------
// K1: opd[b][pix] = sum_k zc[b][k] * M[k][pix],  M = [zernike_maps ; S_mat]
// one thread per pixel, 32 batch accumulators in registers
// ---------------------------------------------------------------------------
__global__ void __launch_bounds__(256)
k1_opd(const float* __restrict__ zern,
       const float* __restrict__ smat,
       float* __restrict__ ws) {
  __shared__ float zc[BATCHN*NK];
  int tid = threadIdx.x;
  for (int i = tid; i < BATCHN*NK; i += blockDim.x) zc[i] = ws[OFF_ZC + i];
  __syncthreads();
  int pix = blockIdx.x*blockDim.x + tid;
  float acc[BATCHN];
  #pragma unroll
  for (int b = 0; b < BATCHN; ++b) acc[b] = 0.0f;
  for (int k = 0; k < NK; ++k) {
    float m = (k < NZT) ? zern[k*OPD*OPD + pix] : smat[(k-NZT)*OPD*OPD + pix];
    #pragma unroll
    for (int b = 0; b < BATCHN; ++b) acc[b] += zc[b*NK + k] * m;
  }
  float* opd = ws + OFF_OPD;
  #pragma unroll
  for (int b = 0; b < BATCHN; ++b) opd[b*OPD*OPD + pix] = acc[b];
}

// ---------------------------------------------------------------------------
// K2: per-bin pruned-DFT twiddles W[j][t][y] = exp(-2*pi*i*(t-48)*y/pN_j)
// integer modulo reduction keeps angles in (-2pi, 2pi)
// ---------------------------------------------------------------------------
__global__ void k2_twiddle(PN pns, float* __restrict__ ws) {
  int gid = blockIdx.x*blockDim.x + threadIdx.x;
  if (gid >= NB*CROP*OPD) return;
  int y = gid & (OPD-1);
  int t = (gid >> 8) % CROP;
  int j = gid / (CROP*OPD);
  int pn = pns.pn[j];
  int m = ((t - 48) * y) % pn;
  float ang = -2.0f * (float)M_PI * (float)m / (float)pn;
  ws[OFF_WR + gid] = cosf(ang);
  ws[OFF_WI + gid] = sinf(ang);
}

// ---------------------------------------------------------------------------
// K3 (per bin): stage-1 complex GEMM  T(96x256) = W(96x256) * P(256x256)
// grid (16 x-tiles, 32 stars), 6 waves/WG; pupil slab in LDS; f32 WMMA
// ---------------------------------------------------------------------------
__global__ void __launch_bounds__(192)
k3_stage1(int j, const float* __restrict__ packed,
          const float* __restrict__ obsc,
          float* __restrict__ ws) {
  __shared__ float sPr[OPD*16];
  __shared__ float sPi[OPD*16];
  int b   = blockIdx.y;
  int x0  = blockIdx.x * 16;
  int tid = threadIdx.x;
  float lam  = packed[(b*NB + j)*3 + 1];
  float cphi = 2.0f*(float)M_PI / lam;
  const float* opd = ws + OFF_OPD + b*OPD*OPD;
  for (int i = tid; i < OPD*16; i += 192) {
    int y = i >> 4, xl = i & 15;
    float o = obsc[y*OPD + x0 + xl];
    float s, c;
    __sincosf(cphi * opd[y*OPD + x0 + xl], &s, &c);
    sPr[i] = o * c;
    sPi[i] = o * s;
  }
  __syncthreads();

  int wave = tid >> 5;
  int lane = tid & 31;
  int lo = lane & 15, hi = lane >> 4;
  const float* Wr = ws + OFF_WR + j*CROP*OPD;
  const float* Wi = ws + OFF_WI + j*CROP*OPD;
  int arow = (wave*16 + lo) * OPD;

  v8f rr = {}, ii = {}, ri = {}, ir = {};
  for (int ks = 0; ks < OPD/4; ++ks) {
    int kb = 4*ks + 2*hi;
    v2f ar = *(const v2f*)(Wr + arow + kb);
    v2f ai = *(const v2f*)(Wi + arow + kb);
    v2f br = { sPr[(kb+0)*16 + lo], sPr[(kb+1)*16 + lo] };
    v2f bi = { sPi[(kb+0)*16 + lo], sPi[(kb+1)*16 + lo] };
    rr = __builtin_amdgcn_wmma_f32_16x16x4_f32(false, ar, false, br, (short)0, rr, false, false);
    ii = __builtin_amdgcn_wmma_f32_16x16x4_f32(false, ai, false, bi, (short)0, ii, false, false);
    ri = __builtin_amdgcn_wmma_f32_16x16x4_f32(false, ar, false, bi, (short)0, ri, false, false);
    ir = __builtin_amdgcn_wmma_f32_16x16x4_f32(false, ai, false, br, (short)0, ir, false, false);
  }
  v8f tr = rr - ii;
  v8f ti = ri + ir;
  float* Tr = ws + OFF_TR + b*CROP*OPD;
  float* Ti = ws + OFF_TI + b*CROP*OPD;
  #pragma unroll
  for (int v = 0; v < 8; ++v) {
    int t = wave*16 + v + 8*hi;
    Tr[t*OPD + x0 + lo] = tr[v];
    Ti[t*OPD + x0 + lo] = ti[v];
  }
}

// ---------------------------------------------------------------------------
// K4 (per bin): stage-2 complex GEMM amp(96x96) = T * W^T, then power |amp|^2
// grid (6 n-tiles, 32 stars), 6 waves/WG (wave = M-tile) -> 1152 waves/bin
// ---------------------------------------------------------------------------
__global__ void __launch_bounds__(192)
k4_stage2(int j, float* __restrict__ ws) {
  int nt  = blockIdx.x;
  int b   = blockIdx.y;
  int tid = threadIdx.x;
  int wave = tid >> 5, lane = tid & 31;
  int lo = lane & 15, hi = lane >> 4;
  const float* Tr = ws + OFF_TR + b*CROP*OPD;
  const float* Ti = ws + OFF_TI + b*CROP*OPD;
  const float* Wr = ws + OFF_WR + j*CROP*OPD;
  const float* Wi = ws + OFF_WI + j*CROP*OPD;
  float* ps = ws + OFF_PS + b*CROP*CROP;
  int arow = (wave*16 + lo) * OPD;
  int brow = (nt*16 + lo) * OPD;
  v8f rr = {}, ii = {}, ri = {}, ir = {};
  for (int ks = 0; ks < OPD/4; ++ks) {
    int kb = 4*ks + 2*hi;
    v2f ar = *(const v2f*)(Tr + arow + kb);
    v2f ai = *(const v2f*)(Ti + arow + kb);
    v2f br = *(const v2f*)(Wr + brow + kb);
    v2f bi = *(const v2f*)(Wi + brow + kb);
    rr = __builtin_amdgcn_wmma_f32_16x16x4_f32(false, ar, false, br, (short)0, rr, false, false);
    ii = __builtin_amdgcn_wmma_f32_16x16x4_f32(false, ai, false, bi, (short)0, ii, false, false);
    ri = __builtin_amdgcn_wmma_f32_16x16x4_f32(false, ar, false, bi, (short)0, ri, false, false);
    ir = __builtin_amdgcn_wmma_f32_16x16x4_f32(false, ai, false, br, (short)0, ir, false, false);
  }
  v8f ampr = rr - ii;
  v8f ampi = ri + ir;
  #pragma unroll
  for (int v = 0; v < 8; ++v) {
    int t = wave*16 + v + 8*hi;
    ps[t*CROP + nt*16 + lo] = ampr[v]*ampr[v] + ampi[v]*ampi[v];
  }
}

// ---------------------------------------------------------------------------
// K5 (per bin): 3x3 box downsample -> 32x32, normalize by image sum,
// SED-weighted accumulate into output (write on j==0)
// ---------------------------------------------------------------------------
__global__ void __launch_bounds__(1024)
k5_accum(int j, const float* __restrict__ packed,
         const float* __restrict__ ws, float* __restrict__ out) {
  __shared__ float red[1024];
  int b   = blockIdx.x;
  int tid = threadIdx.x;
  const float* ps = ws + OFF_PS + b*CROP*CROP;
  int oy = tid >> 5, ox = tid & 31;
  float s = 0.0f;
  #pragma unroll
  for (int dy = 0; dy < 3; ++dy)
    #pragma unroll
    for (int dx = 0; dx < 3; ++dx)
      s += ps[(oy*3 + dy)*CROP + ox*3 + dx];
  s *= (1.0f/9.0f);
  red[tid] = s;
  __syncthreads();
  for (int off = 512; off > 0; off >>= 1) {
    if (tid < off) red[tid] += red[tid + off];
    __syncthreads();
  }
  float total = red[0];
  float sed = packed[(b*NB + j)*3 + 2];
  float val = sed * s / total;
  float* o = out + b*OD*OD + tid;
  if (j == 0) *o = val; else *o += val;
}

// ---------------------------------------------------------------------------
extern "C" void kernel_launch(void* const* d_in, const int* in_sizes, int n_in,
                              void* d_out, int out_size, void* d_ws, size_t ws_size,
                              hipStream_t stream) {
  const float* positions = (const float*)d_in[0];
  const float* packed    = (const float*)d_in[1];
  const float* coeff     = (const float*)d_in[2];
  const float* alpha     = (const float*)d_in[3];
  const float* smat      = (const float*)d_in[4];
  const float* zern      = (const float*)d_in[5];
  const float* obsc      = (const float*)d_in[6];
  const float* obs_pos   = (const float*)d_in[7];
  const float* zprior    = (const float*)d_in[8];
  float* out = (float*)d_out;
  float* ws  = (float*)d_ws;

  PN pns;
  for (int j = 0; j < NB; ++j) {
    double lam = 0.55 + j * (0.35/19.0);
    pns.pn[j] = 2 * (int)llrint(256.0 * 3.0 * lam / (2.0 * 0.55));
  }

  hipLaunchKernelGGL(k0_zcomb, dim3(1), dim3(32), 0, stream,
                     positions, coeff, alpha, obs_pos, zprior, ws);
  hipLaunchKernelGGL(k1_opd, dim3(OPD*OPD/256), dim3(256), 0, stream,
                     zern, smat, ws);
  hipLaunchKernelGGL(k2_twiddle, dim3((NB*CROP*OPD + 255)/256), dim3(256), 0, stream,
                     pns, ws);
  for (int j = 0; j < NB; ++j) {
    hipLaunchKernelGGL(k3_stage1, dim3(16, 32), dim3(192), 0, stream, j, packed, obsc, ws);
    hipLaunchKernelGGL(k4_stage2, dim3(6, 32),  dim3(192), 0, stream, j, ws);
    hipLaunchKernelGGL(k5_accum,  dim3(32),     dim3(1024), 0, stream, j, packed, ws, out);
  }
}